// GMM_64690797412762
// MI455X (gfx1250) — compile-verified
//
#include <hip/hip_runtime.h>
#include <hip/hip_bf16.h>
#include <stdint.h>

typedef __attribute__((ext_vector_type(2))) float v2f;
typedef __attribute__((ext_vector_type(8))) float v8f;

#define KCOMP 16
#define DDIM  64

// ---------------- counter-based RNG: murmur3-finalizer pair + Box-Muller ----
__device__ __forceinline__ uint32_t hash32(uint32_t x) {
  x ^= x >> 16; x *= 0x85EBCA6Bu;
  x ^= x >> 13; x *= 0xC2B2AE35u;
  x ^= x >> 16;
  return x;
}

// Two independent standard normals for counter `ctr` (consecutive K pair).
__device__ __forceinline__ v2f gauss_pair(uint32_t ctr) {
  uint32_t u1i = hash32(ctr * 0x9E3779B9u + 0x6A09E667u) | 1u;  // avoid zero
  uint32_t u2i = hash32(ctr * 0x85EBCA77u + 0xBB67AE85u);
  float u1 = (float)u1i * 2.3283064365386963e-10f;   // 2^-32 -> (0,1]
  float u2 = (float)u2i * 2.3283064365386963e-10f;
  float r  = __builtin_sqrtf(-2.0f * __logf(u1));
  float a  = 6.283185307179586f * u2;
  v2f o;
  o.x = r * __cosf(a);
  o.y = r * __sinf(a);
  return o;
}

// ---------------- kernel 1: weights->offsets, cov = A^T A/D + eps*I, Cholesky
__global__ void __launch_bounds__(256)
gmm_setup_kernel(const float* __restrict__ covdata,
                 const float* __restrict__ weights,
                 float* __restrict__ Lout,   // [16][64][64]
                 int*   __restrict__ cumout, // [17]
                 int N) {
  __shared__ float Ash[DDIM * DDIM];
  __shared__ float Csh[DDIM * DDIM];
  const int k   = blockIdx.x;
  const int tid = threadIdx.x;

  if (k == 0 && tid == 0) {
    float s = 1e-20f;
    for (int i = 0; i < KCOMP; ++i) s += fabsf(weights[i]);
    int cum = 0;
    cumout[0] = 0;
    for (int i = 0; i < KCOMP; ++i) {
      int cnt = (int)rintf((float)N * fabsf(weights[i]) / s);
      cum += cnt;
      cumout[i + 1] = cum;
    }
  }

  const float* A = covdata + (size_t)k * DDIM * DDIM;
  for (int e = tid; e < DDIM * DDIM; e += 256) Ash[e] = A[e];
  __syncthreads();

  // cov[i][l] = sum_j A[j][i]*A[j][l] / D  (+ eps on diag)
  for (int e = tid; e < DDIM * DDIM; e += 256) {
    int i = e >> 6, l = e & 63;
    float acc = 0.0f;
    #pragma unroll 8
    for (int j = 0; j < DDIM; ++j) acc += Ash[j * DDIM + i] * Ash[j * DDIM + l];
    acc *= (1.0f / 64.0f);
    if (i == l) acc += 1e-20f;
    Csh[e] = acc;
  }
  __syncthreads();

  // in-place lower Cholesky
  for (int j = 0; j < DDIM; ++j) {
    if (tid == 0) Csh[j * DDIM + j] = __builtin_sqrtf(Csh[j * DDIM + j]);
    __syncthreads();
    float invd = 1.0f / Csh[j * DDIM + j];
    if (tid > j && tid < DDIM) Csh[tid * DDIM + j] *= invd;
    __syncthreads();
    for (int e = tid; e < DDIM * DDIM; e += 256) {
      int i = e >> 6, l = e & 63;
      if (i > j && l > j && l <= i)
        Csh[i * DDIM + l] -= Csh[i * DDIM + j] * Csh[l * DDIM + j];
    }
    __syncthreads();
  }

  float* Lk = Lout + (size_t)k * DDIM * DDIM;
  for (int e = tid; e < DDIM * DDIM; e += 256) {
    int i = e >> 6, l = e & 63;
    Lk[e] = (l <= i) ? Csh[e] : 0.0f;
  }
}

// ---------------- kernel 2: y = eps @ L^T + mu, permuted store -------------
__global__ void __launch_bounds__(256)
gmm_sample_kernel(const float* __restrict__ mus,
                  const float* __restrict__ Lmat,
                  const int*   __restrict__ cumin,
                  float* __restrict__ out, int N) {
  __shared__ __align__(16) float Lsh[DDIM * DDIM];
  const int tid  = threadIdx.x;
  const int lane = tid & 31;
  const int wave = tid >> 5;
  const int rb   = blockIdx.x * 128;     // block row base (8 waves * 16 rows)
  const int r0   = rb + wave * 16;       // this wave's 16-row tile base

  // block-uniform component range covered by rows [rb, rb+127]
  const int rlast = min(rb + 127, N - 1);
  int cfirst = 0, clast = 0;
  #pragma unroll
  for (int i = 1; i < KCOMP; ++i) {
    int b = cumin[i];
    cfirst += (rb    >= b);
    clast  += (rlast >= b);
  }

  // A fragments: 16x64 standard normals in WMMA f32 16x16x4 A layout.
  // lane holds M = lane&15; VGPR0/1 hold K = 4*kc + 2*(lane>>4) + {0,1}.
  const int m    = lane & 15;
  const int half = lane >> 4;
  const int mrow = r0 + m;
  v2f afrag[16];
  #pragma unroll
  for (int kc = 0; kc < 16; ++kc) {
    uint32_t ctr = (uint32_t)mrow * 32u + (uint32_t)(kc * 2 + half);
    afrag[kc] = gauss_pair(ctr);
  }

  // Permuted output row bases: independent of c and t -> compute once.
  // Signed int offsets so stores fold into saddr + i32 voffset (scale_offset).
  const bool     pow2 = (N & (N - 1)) == 0;
  const uint32_t mask = (uint32_t)(N - 1);
  int obase[8];
  #pragma unroll
  for (int v = 0; v < 8; ++v) {
    uint32_t row  = (uint32_t)(r0 + v + half * 8);
    uint32_t orow = pow2 ? ((2654435761u * row + 0x3ff79c3du) & mask) : row;
    obase[v] = (int)(orow * (uint32_t)DDIM);
  }

  const int col16 = lane & 15;            // N index within column tile
  const bool block_in_range = (rb + 127 < N);

  for (int c = cfirst; c <= clast; ++c) { // uniform loop (usually 1 iter)
    // stage L_c into LDS with 128-bit moves (4 float4 per thread)
    const float4* Lc4 = (const float4*)(Lmat + (size_t)c * DDIM * DDIM);
    float4* Lsh4 = (float4*)Lsh;
    #pragma unroll
    for (int e = 0; e < 4; ++e) Lsh4[tid + 256 * e] = Lc4[tid + 256 * e];
    __syncthreads();

    const int lo = cumin[c];
    const int hi = (c == KCOMP - 1) ? N : min(cumin[c + 1], N);
    const bool uniform = (cfirst == clast) && block_in_range;  // wave-uniform

    #pragma unroll
    for (int t = 0; t < 4; ++t) {
      const int bcol = t * 16 + col16;    // output column == row of L
      v8f acc = {0.f, 0.f, 0.f, 0.f, 0.f, 0.f, 0.f, 0.f};
      #pragma unroll
      for (int kc = 0; kc < 16; ++kc) {
        // B[j][i] = L[i][j]; lane supplies K = 4*kc + 2*half + {0,1}, N = bcol
        v2f b = *(const v2f*)&Lsh[bcol * DDIM + kc * 4 + half * 2];
        acc = __builtin_amdgcn_wmma_f32_16x16x4_f32(
            false, afrag[kc], false, b, (short)0, acc, false, false);
      }
      const float mu = mus[c * DDIM + bcol];

      if (uniform) {
        // fast path: whole block belongs to component c -> no predication
        #pragma unroll
        for (int v = 0; v < 8; ++v)
          __builtin_nontemporal_store(acc[v] + mu, &out[obase[v] + bcol]);
      } else {
        #pragma unroll
        for (int v = 0; v < 8; ++v) {
          const int row = r0 + v + half * 8;
          if (row >= lo && row < hi)
            __builtin_nontemporal_store(acc[v] + mu, &out[obase[v] + bcol]);
        }
      }
    }
    __syncthreads();
  }
}

// ---------------- host launcher --------------------------------------------
extern "C" void kernel_launch(void* const* d_in, const int* in_sizes, int n_in,
                              void* d_out, int out_size, void* d_ws, size_t ws_size,
                              hipStream_t stream) {
  const float* mus     = (const float*)d_in[0];
  const float* covdata = (const float*)d_in[1];
  const float* weights = (const float*)d_in[2];
  // d_in[3] holds N on device; graph capture forbids reading it, but
  // out_size = N * D is host-visible:
  const int N = out_size / DDIM;

  float* Lws = (float*)d_ws;                                   // 256 KB
  int*   cum = (int*)((char*)d_ws + (size_t)KCOMP * DDIM * DDIM * sizeof(float));

  gmm_setup_kernel<<<KCOMP, 256, 0, stream>>>(covdata, weights, Lws, cum, N);

  const int nblk = (N + 127) / 128;
  gmm_sample_kernel<<<nblk, 256, 0, stream>>>(mus, Lws, cum, (float*)d_out, N);
}